// RVLinearlayer_36421322670457
// MI455X (gfx1250) — compile-verified
//
#include <hip/hip_runtime.h>
#include <hip/hip_bf16.h>
#include <math.h>

// ---------------------------------------------------------------------------
// RVLinearLayer forward (moment-propagating Bayesian linear layer)
//   B = 64, size_in = size_out = 1024, all f32.
// Heavy path: Sigma_out[b] = w_mu^T * sigma_in[b] * w_mu  (two 1024^3 GEMMs
// per batch), done with V_WMMA_F32_16X16X4_F32 (the CDNA5 f32 matrix pipe),
// double-buffered LDS fed by GLOBAL_LOAD_ASYNC_TO_LDS_B128 when available.
// ---------------------------------------------------------------------------

typedef __attribute__((ext_vector_type(2))) float v2f;
typedef __attribute__((ext_vector_type(4))) float v4f;
typedef __attribute__((ext_vector_type(8))) float v8f;
typedef __attribute__((ext_vector_type(4))) int   v4i;

#define N_DIM 1024
#define B_DIM 64

#ifndef __has_builtin
#define __has_builtin(x) 0
#endif
#if __has_builtin(__builtin_amdgcn_global_load_async_to_lds_b128) && \
    __has_builtin(__builtin_amdgcn_s_wait_asynccnt)
#define HAVE_ASYNC_LDS 1
#else
#define HAVE_ASYNC_LDS 0
#endif

#if HAVE_ASYNC_LDS
// builtin signature (from hipcc diagnostic): (v4i AS1*, v4i AS3*, imm, imm)
typedef __attribute__((address_space(1))) v4i* gp_v4i;
typedef __attribute__((address_space(3))) v4i* lp_v4i;
__device__ __forceinline__ void async_cp16(const float* g, float* l) {
    // global -> LDS, 16B, tracked by ASYNCcnt
    __builtin_amdgcn_global_load_async_to_lds_b128((gp_v4i)g, (lp_v4i)l, 0, 0);
}
#endif

// ---------------------------------------------------------------------------
// WMMA f32 GEMM: C[b] = A[b] * B[b]  (M = N = K = 1024; A row-major unless
// AKM, in which case A is stored K-major: A[m][k] = Abase[k*1024 + m]).
// Block tile 128x128, K-tile 32. 256 threads = 8 wave32, wave tile 32x64.
// LDS layouts (all rows 16B-aligned, straight b128 global->LDS copies):
//   !AKM : As[m][k]  stride 36  (A fragment = contiguous pair -> ds_load_b64)
//    AKM : As[k][m]  stride 132
//   both : Bs[k][n]  stride 132
// ADD_DIAG: fuse Sigma_out[b,i,i] += diag_term[b,i] into the epilogue.
// ---------------------------------------------------------------------------
template<bool AKM, bool ADD_DIAG>
__global__ __launch_bounds__(256)
void gemm_wmma_f32(const float* __restrict__ Abase,
                   const float* __restrict__ Bbase,
                   float* __restrict__ Cbase,
                   const float* __restrict__ diag_term,
                   unsigned long long aStride,
                   unsigned long long bStride,
                   unsigned long long cStride)
{
    const int b = blockIdx.z;
    const float* __restrict__ A  = Abase + (size_t)b * aStride;
    const float* __restrict__ Bm = Bbase + (size_t)b * bStride;
    float* __restrict__       C  = Cbase + (size_t)b * cStride;

    const int tid  = threadIdx.x;
    const int lane = tid & 31;
    const int wave = tid >> 5;
    const int wm   = wave >> 1;        // 0..3  (M direction, 32 rows each)
    const int wn   = wave & 1;         // 0..1  (N direction, 64 cols each)
    const int ml   = lane & 15;        // low lane half index
    const int lh   = lane >> 4;        // lane group (0/1)

    const int rowBase = blockIdx.y * 128;
    const int colBase = blockIdx.x * 128;

    // double-buffered tiles; 2*(18432 + 16896) = 70656 B of the WGP's 320 KB
    __shared__ __align__(16) float AsBuf[2][128 * 36];   // !AKM view [m][k]
    __shared__ __align__(16) float BsBuf[2][32 * 132];   // [k][n] (AKM A too)

    const v8f vzero = {0.f, 0.f, 0.f, 0.f, 0.f, 0.f, 0.f, 0.f};
    v8f acc[2][4];
#pragma unroll
    for (int i = 0; i < 2; ++i)
#pragma unroll
        for (int j = 0; j < 4; ++j)
            acc[i][j] = vzero;

    // ---- tile staging (global -> LDS), 16B chunks, fully coalesced ----
    auto stage = [&](int k0, int buf) {
#pragma unroll
        for (int i = 0; i < 4; ++i) {
            int c = tid + i * 256;                       // 0..1023 chunk id
            const float* gA;
            float* lA;
            if (!AKM) {
                int r  = c >> 3;                         // row (0..127)
                int kc = (c & 7) << 2;                   // k, mult of 4
                gA = A + (size_t)(rowBase + r) * N_DIM + k0 + kc;
                lA = &AsBuf[buf][r * 36 + kc];
            } else {
                int kr = c >> 5;                         // k row (0..31)
                int mc = (c & 31) << 2;                  // m, mult of 4
                gA = A + (size_t)(k0 + kr) * N_DIM + rowBase + mc;
                lA = &AsBuf[buf][kr * 132 + mc];
            }
            int kr = c >> 5;
            int nc = (c & 31) << 2;
            const float* gB = Bm + (size_t)(k0 + kr) * N_DIM + colBase + nc;
            float* lB = &BsBuf[buf][kr * 132 + nc];
#if HAVE_ASYNC_LDS
            async_cp16(gA, lA);
            async_cp16(gB, lB);
#else
            *(v4f*)lA = *(const v4f*)gA;
            *(v4f*)lB = *(const v4f*)gB;
#endif
        }
    };

    // ---- 8 WMMA k-steps of 4 over one staged K-tile ----
    auto compute = [&](int buf) {
#pragma unroll
        for (int kk = 0; kk < 32; kk += 4) {
            const int kg = kk + (lh << 1);   // this lane group's k pair base
            v2f a0, a1;
            if (!AKM) {
                a0 = *(const v2f*)&AsBuf[buf][(wm * 32 + ml) * 36 + kg];
                a1 = *(const v2f*)&AsBuf[buf][(wm * 32 + 16 + ml) * 36 + kg];
            } else {
                a0[0] = AsBuf[buf][kg * 132 + wm * 32 + ml];
                a0[1] = AsBuf[buf][(kg + 1) * 132 + wm * 32 + ml];
                a1[0] = AsBuf[buf][kg * 132 + wm * 32 + 16 + ml];
                a1[1] = AsBuf[buf][(kg + 1) * 132 + wm * 32 + 16 + ml];
            }
            v2f bf[4];
#pragma unroll
            for (int tn = 0; tn < 4; ++tn) {
                int n = wn * 64 + tn * 16 + ml;
                bf[tn][0] = BsBuf[buf][kg * 132 + n];
                bf[tn][1] = BsBuf[buf][(kg + 1) * 132 + n];
            }
#pragma unroll
            for (int tn = 0; tn < 4; ++tn) {
                acc[0][tn] = __builtin_amdgcn_wmma_f32_16x16x4_f32(
                    false, a0, false, bf[tn], (short)0, acc[0][tn], false, false);
                acc[1][tn] = __builtin_amdgcn_wmma_f32_16x16x4_f32(
                    false, a1, false, bf[tn], (short)0, acc[1][tn], false, false);
            }
        }
    };

    const int nT = N_DIM / 32;
#if HAVE_ASYNC_LDS
    // software pipeline: async engine fills buf^1 while WMMA runs on buf
    stage(0, 0);
    __builtin_amdgcn_s_wait_asynccnt(0);
    __syncthreads();
    for (int t = 0; t < nT; ++t) {
        const int cur = t & 1;
        if (t + 1 < nT) stage((t + 1) * 32, cur ^ 1);
        compute(cur);
        if (t + 1 < nT) __builtin_amdgcn_s_wait_asynccnt(0);
        __syncthreads();
    }
#else
    for (int t = 0; t < nT; ++t) {
        stage(t * 32, 0);
        __syncthreads();
        // prefetch next K-tile into L2 while WMMAs run (global_prefetch_b8)
        if (t + 1 < nT) {
            int k0 = (t + 1) * 32;
            __builtin_prefetch(Bm + (size_t)(k0 + (tid >> 3)) * N_DIM + colBase + (tid & 7) * 16, 0, 1);
            if (!AKM)
                __builtin_prefetch(A + (size_t)(rowBase + (tid >> 1)) * N_DIM + k0 + (tid & 1) * 16, 0, 1);
            else
                __builtin_prefetch(A + (size_t)(k0 + (tid >> 3)) * N_DIM + rowBase + (tid & 7) * 16, 0, 1);
        }
        compute(0);
        __syncthreads();
    }
#endif

    // ---- epilogue: C/D layout -> row = v + 8*(lane>>4), col = lane&15 ----
#pragma unroll
    for (int mt = 0; mt < 2; ++mt)
#pragma unroll
        for (int tn = 0; tn < 4; ++tn)
#pragma unroll
            for (int v = 0; v < 8; ++v) {
                int r  = rowBase + wm * 32 + mt * 16 + v + (lh << 3);
                int cc = colBase + wn * 64 + tn * 16 + ml;
                float val = acc[mt][tn][v];
                if (ADD_DIAG) {
                    if (r == cc) val += diag_term[(size_t)b * N_DIM + r];
                }
                C[(size_t)r * N_DIM + cc] = val;
            }
}

// ---------------------------------------------------------------------------
// mu_out[b,o] = sum_i w_mu[i,o] * mu_in[b,i] + b_mu[o]
// ---------------------------------------------------------------------------
__global__ __launch_bounds__(256)
void mu_out_kernel(const float* __restrict__ mu_in,
                   const float* __restrict__ w_mu,
                   const float* __restrict__ b_mu,
                   float* __restrict__ mu_out)
{
    const int o = blockIdx.x * blockDim.x + threadIdx.x;  // 0..1023
    const int b = blockIdx.y;
    const float* mu = mu_in + (size_t)b * N_DIM;
    float s = 0.f;
    for (int i = 0; i < N_DIM; ++i)
        s = fmaf(w_mu[(size_t)i * N_DIM + o], mu[i], s);
    mu_out[(size_t)b * N_DIM + o] = s + b_mu[o];
}

// ---------------------------------------------------------------------------
// diag_term[b,o] = softplus(b_sigma[o]) + sum_l softplus(w_sigma[o,l]) *
//                                         (sigma_in[b,l,l] + mu_in[b,l]^2)
// One wave32 per (b,o).
// ---------------------------------------------------------------------------
__global__ __launch_bounds__(256)
void diag_term_kernel(const float* __restrict__ sigma_in,
                      const float* __restrict__ mu_in,
                      const float* __restrict__ w_sigma,
                      const float* __restrict__ b_sigma,
                      float* __restrict__ dt)
{
    const int gwave = (blockIdx.x * 256 + threadIdx.x) >> 5;  // 0..65535
    const int lane  = threadIdx.x & 31;
    const int b = gwave >> 10;
    const int o = gwave & 1023;
    const float* __restrict__ srow = w_sigma + (size_t)o * N_DIM;
    const float* __restrict__ sig  = sigma_in + (size_t)b * N_DIM * N_DIM;
    const float* __restrict__ mu   = mu_in + (size_t)b * N_DIM;
    float s = 0.f;
    for (int l = lane; l < N_DIM; l += 32) {
        float wsig = log1pf(__expf(srow[l]));
        float v = sig[(size_t)l * N_DIM + l] + mu[l] * mu[l];
        s = fmaf(wsig, v, s);
    }
#pragma unroll
    for (int off = 16; off > 0; off >>= 1)
        s += __shfl_down(s, off, 32);
    if (lane == 0)
        dt[(size_t)b * N_DIM + o] = s + log1pf(__expf(b_sigma[o]));
}

// ---------------------------------------------------------------------------
// KL = 0.5 * sum_i ( ||w_mu[:,i]||^2 + sum_l wsig[i,l] - 1024 - logdet_i )
// det_i = prod_l wsig[i,l] in f32 (underflows -> 0 -> clamp -1000, as ref).
// Single block of 1024 threads, thread i owns output column i.
// ---------------------------------------------------------------------------
__global__ __launch_bounds__(1024)
void kl_kernel(const float* __restrict__ w_mu,
               const float* __restrict__ w_sigma,
               float* __restrict__ out_kl)
{
    const int i = threadIdx.x;
    float sq = 0.f, tr = 0.f, det = 1.f;
    for (int k = 0; k < N_DIM; ++k) {
        float wm = w_mu[(size_t)k * N_DIM + i];
        sq = fmaf(wm, wm, sq);
        float ws = log1pf(__expf(w_sigma[(size_t)i * N_DIM + k]));
        tr += ws;
        det *= ws;
    }
    float logdet = (det > 0.f) ? __logf(det) : -1000.f;
    float part = 0.5f * (sq + tr - (float)N_DIM - logdet);

    __shared__ float red[1024];
    red[i] = part;
    __syncthreads();
#pragma unroll
    for (int s = 512; s > 0; s >>= 1) {
        if (i < s) red[i] += red[i + s];
        __syncthreads();
    }
    if (i == 0) *out_kl = red[0];
}

// ---------------------------------------------------------------------------
extern "C" void kernel_launch(void* const* d_in, const int* in_sizes, int n_in,
                              void* d_out, int out_size, void* d_ws, size_t ws_size,
                              hipStream_t stream)
{
    (void)in_sizes; (void)n_in; (void)out_size; (void)ws_size;

    const float* mu_in    = (const float*)d_in[0];  // (64,1024,1)
    const float* sigma_in = (const float*)d_in[1];  // (64,1024,1024)
    const float* w_mu     = (const float*)d_in[2];  // (1024,1024) [in,out]
    const float* w_sigma  = (const float*)d_in[3];  // (1024,1024) [out,in]
    const float* b_mu     = (const float*)d_in[4];  // (1024,1)
    const float* b_sigma  = (const float*)d_in[5];  // (1024,)

    float* out    = (float*)d_out;
    float* mu_out = out;                                         // 64*1024
    float* Sigma  = out + (size_t)B_DIM * N_DIM;                 // 64*1024*1024
    float* kl     = Sigma + (size_t)B_DIM * N_DIM * N_DIM;       // 1

    // workspace: tmp[b] = sigma_in[b] @ w_mu (256 MB) + diag_term (256 KB)
    float* tmp = (float*)d_ws;
    float* dt  = tmp + (size_t)B_DIM * N_DIM * N_DIM;

    const unsigned long long mat = (unsigned long long)N_DIM * N_DIM;

    mu_out_kernel<<<dim3(N_DIM / 256, B_DIM), 256, 0, stream>>>(mu_in, w_mu, b_mu, mu_out);
    diag_term_kernel<<<(B_DIM * N_DIM * 32) / 256, 256, 0, stream>>>(
        sigma_in, mu_in, w_sigma, b_sigma, dt);

    dim3 g(N_DIM / 128, N_DIM / 128, B_DIM);
    // pass 1: tmp[b] = sigma_in[b] (row-major) @ w_mu (row-major)
    gemm_wmma_f32<false, false><<<g, 256, 0, stream>>>(
        sigma_in, w_mu, tmp, nullptr, mat, 0ull, mat);
    // pass 2: Sigma[b] = w_mu^T (K-major view of w_mu) @ tmp[b], + diag fused
    gemm_wmma_f32<true, true><<<g, 256, 0, stream>>>(
        w_mu, tmp, Sigma, dt, 0ull, mat, mat);

    kl_kernel<<<1, 1024, 0, stream>>>(w_mu, w_sigma, kl);
}